// NLIF_2130303779220
// MI455X (gfx1250) — compile-verified
//
#include <hip/hip_runtime.h>

#define NN   2048
#define TT   2048
#define NBLK 128
#define TPB  512
#define NWAVES (TPB / 32)   // 16 waves: 8 for s@Wsyn, 8 for s_fast@Wfast

typedef float v2f __attribute__((ext_vector_type(2)));
typedef float v8f __attribute__((ext_vector_type(8)));

// One K=4 step of the mat-vec as a V_WMMA_F32_16X16X4_F32.
// A (16x4 f32): row 0 = u[k..k+3], rows 1..15 = 0.
//   layout: VGPR0 = {K0 lanes0-15 (M=lane), K2 lanes16-31}; VGPR1 = {K1, K3}
//   -> lane0.a = (u[k],u[k+1]); lane16.a = (u[k+2],u[k+3]); others 0.
// B (4x16 f32): B[kk][n] = W[k+kk][j0+n]
//   layout mirror: VGPR0 = {row k lanes0-15, row k+2 lanes16-31}; VGPR1 = {k+1, k+3}
// D row0 (lanes 0-15, element 0) accumulates the 16 outputs.
__device__ __forceinline__ v8f wmma_k4(const float* __restrict__ W,
                                       const float* __restrict__ uh,
                                       int k, int jn, int n16, int rsel,
                                       bool maskDiag, v8f c)
{
  int r = k + (rsel << 1);                       // this lane's first row: k or k+2
  float bx = W[(size_t)r * NN + jn];             // row r
  float by = W[(size_t)r * NN + NN + jn];        // row r+1
  if (maskDiag) {                                // W_syn * (1 - eye), applied on the fly
    bx = (r     == jn) ? 0.f : bx;
    by = (r + 1 == jn) ? 0.f : by;
  }
  float ax = uh[r];                              // broadcast loads, L0 hits
  float ay = uh[r + 1];
  v2f a, b;
  a.x = (n16 == 0) ? ax : 0.f;
  a.y = (n16 == 0) ? ay : 0.f;
  b.x = bx;
  b.y = by;
  return __builtin_amdgcn_wmma_f32_16x16x4_f32(false, a, false, b, (short)0, c,
                                               false, false);
}

// Device-wide generation barrier (all NBLK blocks resident by construction).
__device__ __forceinline__ void grid_barrier(unsigned* cnt, unsigned* gen, unsigned nb)
{
  __threadfence();                               // release: make our stores device-visible
  __syncthreads();
  if (threadIdx.x == 0) {
    unsigned g    = __hip_atomic_load(gen, __ATOMIC_RELAXED, __HIP_MEMORY_SCOPE_AGENT);
    unsigned prev = __hip_atomic_fetch_add(cnt, 1u, __ATOMIC_ACQ_REL, __HIP_MEMORY_SCOPE_AGENT);
    if (prev == nb - 1u) {
      __hip_atomic_store(cnt, 0u, __ATOMIC_RELAXED, __HIP_MEMORY_SCOPE_AGENT);
      __hip_atomic_fetch_add(gen, 1u, __ATOMIC_RELEASE, __HIP_MEMORY_SCOPE_AGENT);
    } else {
      while (__hip_atomic_load(gen, __ATOMIC_ACQUIRE, __HIP_MEMORY_SCOPE_AGENT) == g)
        __builtin_amdgcn_s_sleep(2);
    }
  }
  __syncthreads();
  __threadfence();                               // acquire: drop stale cached lines
}

__global__ void __launch_bounds__(TPB, 1)
nlif_persistent(const float* __restrict__ x_in,   // (T,2)
                const float* __restrict__ Wsyn,   // (N,N)
                const float* __restrict__ Wfast,  // (N,N)
                const float* __restrict__ Win,    // (2,N)
                const float* __restrict__ Io,     // (N,)
                const float* __restrict__ Omat,   // (2,N)
                float* __restrict__ out,
                float* __restrict__ wsF,
                unsigned* __restrict__ bar)
{
  float* uBuf0 = wsF;                 // [s(2048) | s_fast(2048)], parity 0
  float* uBuf1 = wsF + 2 * NN;        // parity 1
  float* vArr  = wsF + 4 * NN;        // membrane potential
  float* par   = wsF + 5 * NN;        // readout partials: [parity][block][2]
  float* spikes   = out;
  float* readouts = out + (size_t)TT * NN;
  float* vs       = out + (size_t)TT * NN + 2 * TT;

  const int bid  = blockIdx.x;
  const int tid  = threadIdx.x;
  const int wave = tid >> 5;
  const int lane = tid & 31;
  const int n16  = lane & 15;
  const int rsel = lane >> 4;
  const int j0   = bid << 4;          // this block's 16 output neurons
  const int jn   = j0 + n16;
  const int hsel  = wave >> 3;        // 0: s@Wsyn, 1: s_fast@Wfast
  const int kloc0 = (wave & 7) << 8;  // 256-wide K slice inside the half
  const float* Wm = hsel ? Wfast : Wsyn;
  const bool maskDiag = (hsel == 0);

  __shared__ float red[NWAVES * 16];
  __shared__ float snewS[16];
  __shared__ float xb[2];

  for (int t = 0; t < TT; ++t) {
    const int pr = t & 1;
    const float* uCur = pr ? uBuf1 : uBuf0;
    float* uNext      = pr ? uBuf0 : uBuf1;

    // Deterministic fold of previous step's readout partials (post-barrier safe).
    if (bid == 0 && t > 0 && tid < 2) {
      const float* p = par + (pr ^ 1) * (NBLK * 2);
      float acc = 0.f;
      for (int g2 = 0; g2 < NBLK; ++g2) acc += p[g2 * 2 + tid];
      readouts[(size_t)(t - 1) * 2 + tid] = acc;
    }
    if (tid < 2) xb[tid] = x_in[2 * t + tid];

    // ---- WMMA mat-vec: this wave's 256-long K slice, 4 independent accumulators ----
    const float* uh = uCur + hsel * NN;
    v8f c0 = {}, c1 = {}, c2 = {}, c3 = {};
    for (int it = 0; it < 16; ++it) {
      int kb = kloc0 + it * 16;
      c0 = wmma_k4(Wm, uh, kb + 0,  jn, n16, rsel, maskDiag, c0);
      c1 = wmma_k4(Wm, uh, kb + 4,  jn, n16, rsel, maskDiag, c1);
      c2 = wmma_k4(Wm, uh, kb + 8,  jn, n16, rsel, maskDiag, c2);
      c3 = wmma_k4(Wm, uh, kb + 12, jn, n16, rsel, maskDiag, c3);
    }
    float partial = c0[0] + c1[0] + c2[0] + c3[0];   // D row 0 lives in lanes 0-15
    if (lane < 16) red[wave * 16 + n16] = partial;
    __syncthreads();

    // ---- element-wise neuron update for this block's 16 neurons ----
    if (tid < 16) {
      const int j = j0 + tid;
      float acc = 0.f;
      #pragma unroll
      for (int w = 0; w < NWAVES; ++w) acc += red[w * 16 + tid];
      float Itot = acc + xb[0] * Win[j] + xb[1] * Win[NN + j] + Io[j];
      float vOld = vArr[j];
      float sOld = uCur[j];
      float dv   = (Itot - vOld) / 10.0f;
      float vN   = vOld + dv;
      float gat  = fminf(fmaxf(vN, -1.f), 1.f);
      float dvc  = fminf(fmaxf(dv, -1.f), 1.f);
      float gd   = gat * dvc;
      float sNew = sOld + (gd - sOld) / 10.0f;
      float sp   = (vN >=  1.f) ? 1.f : 0.f;
      float sn   = (vN <= -1.f) ? 1.f : 0.f;
      float vNew = (1.f + sp - sn) * vN;   // exact reference arithmetic (incl. its quirk)
      vArr[j]       = vNew;
      uNext[j]      = sNew;
      uNext[NN + j] = gd;                  // s_fast_new
      spikes[(size_t)t * NN + j] = sp;
      vs[(size_t)t * NN + j]     = vNew;
      snewS[tid] = sNew;
    }
    __syncthreads();

    // per-block readout partial: sum_m O[r][j0+m] * s_new[j0+m]
    if (tid < 2) {
      const float* Or = Omat + tid * NN + j0;
      float acc = 0.f;
      #pragma unroll
      for (int m = 0; m < 16; ++m) acc += Or[m] * snewS[m];
      par[pr * (NBLK * 2) + bid * 2 + tid] = acc;
    }

    grid_barrier(bar, bar + 1, NBLK);
  }

  // last step's readout
  if (bid == 0 && tid < 2) {
    const float* p = par + ((TT - 1) & 1) * (NBLK * 2);
    float acc = 0.f;
    for (int g2 = 0; g2 < NBLK; ++g2) acc += p[g2 * 2 + tid];
    readouts[(size_t)(TT - 1) * 2 + tid] = acc;
  }
}

extern "C" void kernel_launch(void* const* d_in, const int* in_sizes, int n_in,
                              void* d_out, int out_size, void* d_ws, size_t ws_size,
                              hipStream_t stream)
{
  const float* x_in  = (const float*)d_in[0];
  const float* Wsyn  = (const float*)d_in[1];
  const float* Wfast = (const float*)d_in[2];
  const float* Win   = (const float*)d_in[3];
  const float* Io    = (const float*)d_in[4];
  const float* Omat  = (const float*)d_in[5];

  unsigned* bar = (unsigned*)d_ws;                       // [count, gen]
  float* wsF    = (float*)((char*)d_ws + 256);
  // zero: barrier + u double-buffer + v + readout partials (also resets 0xAA poison,
  // and makes every call start from identical state -> deterministic)
  size_t wsBytes = 256 + (size_t)(5 * NN + 2 * NBLK * 2) * sizeof(float);
  hipMemsetAsync(d_ws, 0, wsBytes, stream);

  hipLaunchKernelGGL(nlif_persistent, dim3(NBLK), dim3(TPB), 0, stream,
                     x_in, Wsyn, Wfast, Win, Io, Omat,
                     (float*)d_out, wsF, bar);
}